// ToRGBLayer_63995012711095
// MI455X (gfx1250) — compile-verified
//
#include <hip/hip_runtime.h>

typedef __attribute__((ext_vector_type(2))) float v2f;
typedef __attribute__((ext_vector_type(4))) float v4f;
typedef __attribute__((ext_vector_type(8))) float v8f;

#define BATCH   8
#define CIN     512
#define COUT    3
#define WDIM    512
#define HW      65536          // 256*256
#define CLAMPV  256.0f
#define SLDS_STRIDE 516        // 512 + 4 pad -> LDS bank rotation, conflict-free A loads

// ---------------------------------------------------------------------------
// Kernel 1: styles[b][c] = (dot(w[b,:], affine_w[c,:]) / sqrt(WDIM) + affine_b[c]) / sqrt(CIN)
// Tiny (8x512 out, 2M MACs) -- one block per batch row, w[b] staged in LDS.
// ---------------------------------------------------------------------------
__global__ void __launch_bounds__(512)
styles_kernel(const float* __restrict__ w,
              const float* __restrict__ affine_w,
              const float* __restrict__ affine_b,
              float* __restrict__ styles)
{
    __shared__ float wl[WDIM];
    const int b = blockIdx.x;
    const int c = threadIdx.x;          // 0..511, one output channel per thread

    wl[c] = w[b * WDIM + c];
    __syncthreads();

    const float inv_sqrt_wdim = 0.04419417382415922f;   // 1/sqrt(512)
    const float inv_sqrt_cin  = 0.04419417382415922f;   // 1/sqrt(512)

    const v4f* arow = (const v4f*)(affine_w + (size_t)c * WDIM);
    const v4f* wv   = (const v4f*)wl;

    float acc = 0.0f;
#pragma unroll 4
    for (int k = 0; k < WDIM / 4; ++k) {
        v4f a  = arow[k];
        v4f ww = wv[k];
        acc += a.x * ww.x + a.y * ww.y + a.z * ww.z + a.w * ww.w;
    }
    styles[b * CIN + c] = (acc * inv_sqrt_wdim + affine_b[c]) * inv_sqrt_cin;
}

// ---------------------------------------------------------------------------
// Kernel 2: out[b,o,p] = clamp( sum_c (weight[o,c]*styles[b,c]) * x[b,c,p] + bias[o] )
// Pure HBM-stream of x (1.07 GB); reduction on the matrix pipe via
// V_WMMA_F32_16X16X4_F32. M = output channel (3 of 16 rows live),
// N = 16 pixels per wave, K-loop over 512 channels in steps of 4.
// ---------------------------------------------------------------------------
__global__ void __launch_bounds__(512)
torgb_kernel(const float* __restrict__ x,
             const float* __restrict__ weight,
             const float* __restrict__ bias,
             const float* __restrict__ styles,
             float* __restrict__ out)
{
    __shared__ float s_lds[16 * SLDS_STRIDE];   // 33 KB scale table: s[m][c]

    const int tid      = threadIdx.x;           // 0..511 (16 waves of 32)
    const int b        = blockIdx.x >> 8;       // 256 blocks per batch image
    const int blk_in_b = blockIdx.x & 255;

    // Build combined scale table once per block (rows 3..15 zero-padded).
    for (int idx = tid; idx < 16 * CIN; idx += 512) {
        const int m = idx >> 9;                 // / 512
        const int c = idx & (CIN - 1);
        float v = 0.0f;
        if (m < COUT) v = weight[m * CIN + c] * styles[b * CIN + c];
        s_lds[m * SLDS_STRIDE + c] = v;
    }
    __syncthreads();

    const int wave  = tid >> 5;
    const int lane  = tid & 31;
    const int lcol  = lane & 15;                // N index / A-matrix row M
    const int lhalf = (lane < 16) ? 0 : 2;      // K sub-offset per lane half
    const int p0    = blk_in_b * 256 + wave * 16;

    // Per-lane streaming pointer into x: loads at +0 and +HW, bump by 4*HW/iter.
    const float* xb = x + (size_t)b * CIN * HW + (size_t)lhalf * HW + p0 + lcol;
    const float* sldp = &s_lds[lcol * SLDS_STRIDE + lhalf];

    v8f acc = {};
#pragma unroll 4
    for (int c0 = 0; c0 < CIN; c0 += 4) {
        // A: s[row=lcol][c0+lhalf .. +1]  (ds_load_b64, bank-rotated)
        v2f a = *(const v2f*)(sldp + c0);
        // B: x[c0+lhalf][p] , x[c0+lhalf+1][p]  (two coalesced global_load_b32)
        v2f bb;
        bb.x = xb[0];
        bb.y = xb[HW];
        xb += 4 * HW;
        // D = A(16x4) * B(4x16) + C   -> v_wmma_f32_16x16x4_f32
        acc = __builtin_amdgcn_wmma_f32_16x16x4_f32(
                  false, a, false, bb, (short)0, acc, false, false);
    }

    // Epilogue: lanes 0..15 hold M=0..7 in VGPRs 0..7; rows 0..2 are the RGB outs.
    if (lane < 16) {
        const int p = p0 + lane;
        float* ob = out + (size_t)b * COUT * HW + p;
        float v0 = fminf(fmaxf(acc[0] + bias[0], -CLAMPV), CLAMPV);
        float v1 = fminf(fmaxf(acc[1] + bias[1], -CLAMPV), CLAMPV);
        float v2 = fminf(fmaxf(acc[2] + bias[2], -CLAMPV), CLAMPV);
        ob[0 * HW] = v0;
        ob[1 * HW] = v1;
        ob[2 * HW] = v2;
    }
}

// ---------------------------------------------------------------------------
extern "C" void kernel_launch(void* const* d_in, const int* in_sizes, int n_in,
                              void* d_out, int out_size, void* d_ws, size_t ws_size,
                              hipStream_t stream)
{
    const float* x        = (const float*)d_in[0];   // (8,512,256,256)
    const float* w        = (const float*)d_in[1];   // (8,512)
    const float* weight   = (const float*)d_in[2];   // (3,512,1,1)
    const float* bias     = (const float*)d_in[3];   // (3,)
    const float* affine_w = (const float*)d_in[4];   // (512,512)
    const float* affine_b = (const float*)d_in[5];   // (512,)
    float*       out      = (float*)d_out;           // (8,3,256,256)
    float*       styles   = (float*)d_ws;            // 8*512 floats scratch

    styles_kernel<<<BATCH, 512, 0, stream>>>(w, affine_w, affine_b, styles);
    torgb_kernel<<<BATCH * 256, 512, 0, stream>>>(x, weight, bias, styles, out);
}